// LennardJonesEnergy_57526791962696
// MI455X (gfx1250) — compile-verified
//
#include <hip/hip_runtime.h>
#include <hip/hip_bf16.h>

typedef __attribute__((ext_vector_type(2))) float v2f;
typedef __attribute__((ext_vector_type(8))) float v8f;

#define B_BATCH 32
#define NPART   1024
#define THREADS 256

// ---------------------------------------------------------------------------
// Kernel 1: per-batch mean and oscillator energy.
//   wsMean[b*3+c] = mean of coordinate c for batch b
//   wsOsc[b]      = 0.25 * 0.5 * sum_i |x_i - mean|^2
// ---------------------------------------------------------------------------
__global__ void lj_batch_stats(const float* __restrict__ pos,
                               float* __restrict__ wsMean,
                               float* __restrict__ wsOsc) {
    __shared__ float rx[THREADS], ry[THREADS], rz[THREADS];
    __shared__ float meanS[3];
    const int b = blockIdx.x;
    const int tid = threadIdx.x;
    const float* P = pos + (size_t)b * NPART * 3;

    float sx = 0.f, sy = 0.f, sz = 0.f;
    for (int p = tid; p < NPART; p += THREADS) {
        sx += P[3 * p + 0];
        sy += P[3 * p + 1];
        sz += P[3 * p + 2];
    }
    rx[tid] = sx; ry[tid] = sy; rz[tid] = sz;
    __syncthreads();
    for (int s = THREADS / 2; s > 0; s >>= 1) {
        if (tid < s) {
            rx[tid] += rx[tid + s];
            ry[tid] += ry[tid + s];
            rz[tid] += rz[tid + s];
        }
        __syncthreads();
    }
    if (tid == 0) {
        meanS[0] = rx[0] * (1.0f / NPART);
        meanS[1] = ry[0] * (1.0f / NPART);
        meanS[2] = rz[0] * (1.0f / NPART);
    }
    __syncthreads();
    const float mx = meanS[0], my = meanS[1], mz = meanS[2];

    float ss = 0.f;
    for (int p = tid; p < NPART; p += THREADS) {
        float dx = P[3 * p + 0] - mx;
        float dy = P[3 * p + 1] - my;
        float dz = P[3 * p + 2] - mz;
        ss = fmaf(dx, dx, ss);
        ss = fmaf(dy, dy, ss);
        ss = fmaf(dz, dz, ss);
    }
    rx[tid] = ss;
    __syncthreads();
    for (int s = THREADS / 2; s > 0; s >>= 1) {
        if (tid < s) rx[tid] += rx[tid + s];
        __syncthreads();
    }
    if (tid == 0) {
        wsMean[b * 3 + 0] = mx;
        wsMean[b * 3 + 1] = my;
        wsMean[b * 3 + 2] = mz;
        wsOsc[b] = 0.125f * rx[0];   // 0.25 * 0.5
    }
}

__device__ __forceinline__ float wave_sum16(float v) {
    // Butterfly over a 16-lane group (xor masks stay inside the group).
    v += __shfl_xor(v, 1, 32);
    v += __shfl_xor(v, 2, 32);
    v += __shfl_xor(v, 4, 32);
    v += __shfl_xor(v, 8, 32);
    return v;
}

// One 16x16 pair tile: WMMA Gram block + LJ nonlinearity + accumulation.
// MASKED=true only for the single tile that can contain the i==j diagonal.
template <bool MASKED>
__device__ __forceinline__ void lj_tile(int hi, v2f Av, float4 pj, int jrel,
                                        const float (&ni)[8], float (&S)[8],
                                        float (&Fx)[8], float (&Fy)[8],
                                        float (&Fz)[8], float& E0, float& E1) {
    v2f Bv;
    Bv.x = hi ? pj.z : pj.x;
    Bv.y = hi ? 0.0f : pj.y;
    v8f Cz = {};
    // D[r] = dot(x_{row_base+r}, x_{j0+l16})
    v8f D = __builtin_amdgcn_wmma_f32_16x16x4_f32(
        false, Av, false, Bv, (short)0, Cz, false, false);
    const float nj = pj.w;
#pragma unroll
    for (int r = 0; r < 8; ++r) {
        float sq   = ni[r] + fmaf(-2.0f, D[r], nj);
        float inv2 = __builtin_amdgcn_rcpf(sq);
        if (MASKED) inv2 = (jrel == r) ? 0.0f : inv2;   // mask i==j
        float inv6 = inv2 * inv2 * inv2;
        // energy: inv12 - 2*inv6 = inv6*(inv6 - 2); split accumulators
        if (r & 1) E1 = fmaf(inv6, inv6 - 2.0f, E1);
        else       E0 = fmaf(inv6, inv6 - 2.0f, E0);
        // force coefficient / 12: (inv6^2 - inv6) * inv2
        float craw = fmaf(inv6, inv6, -inv6) * inv2;
        S[r]  += craw;
        Fx[r] = fmaf(craw, pj.x, Fx[r]);
        Fy[r] = fmaf(craw, pj.y, Fy[r]);
        Fz[r] = fmaf(craw, pj.z, Fz[r]);
    }
}

// ---------------------------------------------------------------------------
// Kernel 2: main all-pairs LJ kernel.
// Grid: 256 blocks = 32 batches x 8 i-tiles of 128 rows. 8 wave32 per block,
// each wave owns 16 i-rows and streams all 1024 j in 16-wide tiles.
// The single diagonal-containing tile is peeled (masked); the other 63 run
// mask-free with the next tile's LDS load prefetched ahead of the math.
// ---------------------------------------------------------------------------
__global__ void lj_main(const float* __restrict__ pos,
                        const float* __restrict__ wsMean,
                        float* __restrict__ wsPair,
                        float* __restrict__ outF) {
    __shared__ float4 sP[NPART];    // (x, y, z, |x|^2)  -> 16 KB
    __shared__ float  eShare[8];

    const int tid   = threadIdx.x;
    const int b     = blockIdx.x >> 3;
    const int itile = blockIdx.x & 7;
    const int wave  = tid >> 5;
    const int lane  = tid & 31;
    const int l16   = lane & 15;
    const int hi    = (lane >> 4) & 1;

    // Stage batch positions + norms into LDS.
    const float* Pb = pos + (size_t)b * NPART * 3;
    for (int p = tid; p < NPART; p += THREADS) {
        float x = Pb[3 * p + 0];
        float y = Pb[3 * p + 1];
        float z = Pb[3 * p + 2];
        sP[p] = make_float4(x, y, z, fmaf(x, x, fmaf(y, y, z * z)));
    }
    __syncthreads();

    const int i0       = itile * 128 + wave * 16;  // this wave's 16 i-rows
    const int row_base = i0 + hi * 8;              // rows this lane accumulates

    // A matrix (16x4 f32): lane l16 = row i0+l16; lanes 0-15 hold K={0,1}=(x,y),
    // lanes 16-31 hold K={2,3}=(z,0).
    const float4 piA = sP[i0 + l16];
    v2f Av;
    Av.x = hi ? piA.z : piA.x;
    Av.y = hi ? 0.0f  : piA.y;

    float ni[8];
#pragma unroll
    for (int r = 0; r < 8; ++r) ni[r] = sP[row_base + r].w;

    float S[8], Fx[8], Fy[8], Fz[8];
#pragma unroll
    for (int r = 0; r < 8; ++r) { S[r] = 0.f; Fx[r] = 0.f; Fy[r] = 0.f; Fz[r] = 0.f; }
    float E0 = 0.f, E1 = 0.f;

    const int itileJ = i0 >> 4;   // the one j-tile that contains the diagonal

    // Peeled masked diagonal tile.
    {
        const float4 pjd = sP[i0 + l16];
        lj_tile<true>(hi, Av, pjd, i0 + l16 - row_base,
                      ni, S, Fx, Fy, Fz, E0, E1);
    }

    // Remaining 63 tiles, mask-free, with software-pipelined LDS prefetch.
    const int t0 = (itileJ == 0) ? 1 : 0;
    int tcur = t0;
    float4 pj = sP[(t0 << 4) + l16];
    for (int k = 0; k < 63; ++k) {
        int tnext = tcur + 1;
        if (tnext == itileJ) ++tnext;    // skip the diagonal tile
        if (tnext > 63) tnext = t0;      // last iteration: dummy prefetch
        const float4 pjn = sP[(tnext << 4) + l16];
        lj_tile<false>(hi, Av, pj, 0, ni, S, Fx, Fy, Fz, E0, E1);
        pj = pjn;
        tcur = tnext;
    }

    // Row reductions across the 16 lanes sharing each row.
#pragma unroll
    for (int r = 0; r < 8; ++r) {
        S[r]  = wave_sum16(S[r]);
        Fx[r] = wave_sum16(Fx[r]);
        Fy[r] = wave_sum16(Fy[r]);
        Fz[r] = wave_sum16(Fz[r]);
    }

    const float mx = wsMean[b * 3 + 0];
    const float my = wsMean[b * 3 + 1];
    const float mz = wsMean[b * 3 + 2];

    // F_i = 12*(S_i * x_i - sum_j c x_j) - 0.25*(x_i - mean)
#pragma unroll
    for (int r = 0; r < 8; ++r) {
        if (l16 == r) {
            const int i = row_base + r;
            const float4 p = sP[i];
            float fx = 12.0f * fmaf(S[r], p.x, -Fx[r]) - 0.25f * (p.x - mx);
            float fy = 12.0f * fmaf(S[r], p.y, -Fy[r]) - 0.25f * (p.y - my);
            float fz = 12.0f * fmaf(S[r], p.z, -Fz[r]) - 0.25f * (p.z - mz);
            float* o = outF + (size_t)(b * NPART + i) * 3;
            o[0] = fx; o[1] = fy; o[2] = fz;
        }
    }

    // Energy: full-wave butterfly, then per-block LDS reduce (deterministic).
    float E = E0 + E1;
    E += __shfl_xor(E, 1, 32);
    E += __shfl_xor(E, 2, 32);
    E += __shfl_xor(E, 4, 32);
    E += __shfl_xor(E, 8, 32);
    E += __shfl_xor(E, 16, 32);
    if (lane == 0) eShare[wave] = E;
    __syncthreads();
    if (tid == 0) {
        float s = 0.f;
        for (int w = 0; w < 8; ++w) s += eShare[w];
        wsPair[blockIdx.x] = 0.5f * s;   // i!=j counted twice
    }
}

// ---------------------------------------------------------------------------
// Kernel 3: deterministic energy finalize.
// ---------------------------------------------------------------------------
__global__ void lj_finalize(const float* __restrict__ wsOsc,
                            const float* __restrict__ wsPair,
                            float* __restrict__ outE) {
    int b = threadIdx.x;
    if (b < B_BATCH) {
        float s = wsOsc[b];
        for (int k = 0; k < 8; ++k) s += wsPair[b * 8 + k];
        outE[b] = s;
    }
}

extern "C" void kernel_launch(void* const* d_in, const int* in_sizes, int n_in,
                              void* d_out, int out_size, void* d_ws, size_t ws_size,
                              hipStream_t stream) {
    const float* pos = (const float*)d_in[0];   // [32, 1024, 3] fp32
    float* out = (float*)d_out;                 // [32] energy ++ [32*1024*3] forces
    float* ws = (float*)d_ws;
    float* wsMean = ws;          // 96 floats
    float* wsOsc  = ws + 96;     // 32 floats
    float* wsPair = ws + 128;    // 256 floats

    lj_batch_stats<<<B_BATCH, THREADS, 0, stream>>>(pos, wsMean, wsOsc);
    lj_main<<<B_BATCH * 8, THREADS, 0, stream>>>(pos, wsMean, wsPair, out + B_BATCH);
    lj_finalize<<<1, 32, 0, stream>>>(wsOsc, wsPair, out);
}